// PagedAttentionWithALiBi_40157944217908
// MI455X (gfx1250) — compile-verified
//
#include <hip/hip_runtime.h>

#define NUM_SEQS   32
#define N_HEAD     16
#define HEAD_DIM   128
#define BLOCK_SZ   16
#define MAX_BLOCKS 128
#define MAX_CTX    (MAX_BLOCKS * BLOCK_SZ)

#define WAVES    8
#define NTHREADS (WAVES * 32)

typedef float v2f __attribute__((ext_vector_type(2)));
typedef float v4f __attribute__((ext_vector_type(4)));
typedef float v8f __attribute__((ext_vector_type(8)));

__global__ __launch_bounds__(NTHREADS) void paged_attn_alibi_wmma(
    const float* __restrict__ q,
    const float* __restrict__ k_cache,
    const float* __restrict__ v_cache,
    const int*   __restrict__ block_tables,
    const int*   __restrict__ context_lens,
    float*       __restrict__ out)
{
    const int h    = blockIdx.x;          // head
    const int s    = blockIdx.y;          // sequence
    const int tid  = threadIdx.x;
    const int wave = tid >> 5;
    const int lane = tid & 31;
    const int hi   = lane >> 4;           // 0: lanes 0-15, 1: lanes 16-31
    const int lo   = lane & 15;           // key index within KV block

    __shared__ __align__(16) float qs[HEAD_DIM];
    __shared__ float red_m[WAVES];
    __shared__ float red_l[WAVES];
    __shared__ __align__(16) float red_acc[WAVES][HEAD_DIM];

    const float scale = 0.08838834764831845f;            // 1/sqrt(128)
    const float slope = exp2f(-0.5f * (float)(h + 1));   // ALiBi slope 2^(-8(h+1)/16)

    int ctx = context_lens[s];
    ctx = ctx < 1 ? 1 : (ctx > MAX_CTX ? MAX_CTX : ctx);
    const int qpos    = ctx - 1;
    const int nblocks = (ctx + BLOCK_SZ - 1) / BLOCK_SZ;

    // Stage q[s,h,:] into LDS (broadcast source for WMMA A operand).
    if (tid < HEAD_DIM) qs[tid] = q[((size_t)s * N_HEAD + h) * HEAD_DIM + tid];
    __syncthreads();

    float run_m = -INFINITY;   // running max
    float run_l = 0.0f;        // running exp-sum
    v4f   acc   = {0.0f, 0.0f, 0.0f, 0.0f};   // lane owns out dims [4*lane, 4*lane+4)

    const int* btrow = block_tables + (size_t)s * MAX_BLOCKS;

    for (int blk = wave; blk < nblocks; blk += WAVES) {
        const int    pb     = btrow[blk];
        const size_t kvbase = ((size_t)pb * N_HEAD + h) * (BLOCK_SZ * HEAD_DIM);

        // Lane supplies K[key=lo][4c + 2*hi .. +1] per chunk (ISA A/B k-layout:
        // lanes 0-15 hold k={0,1}, lanes 16-31 hold k={2,3}).
        const float* krow = k_cache + kvbase + (size_t)lo * HEAD_DIM + 2 * hi;

        // ---- scores(16 keys) = K_block · q  via V_WMMA_F32_16X16X4_F32 ----
        v8f c0 = {0, 0, 0, 0, 0, 0, 0, 0};
        v8f c1 = {0, 0, 0, 0, 0, 0, 0, 0};
#pragma unroll
        for (int c = 0; c < 32; c += 2) {
            v2f a0 = *(const v2f*)(&qs[4 * c + 2 * hi]);          // A: q broadcast over M
            v2f b0 = *(const v2f*)(krow + 4 * c);                 // B: K^T chunk
            c0 = __builtin_amdgcn_wmma_f32_16x16x4_f32(
                     false, a0, false, b0, (short)0, c0, false, false);
            v2f a1 = *(const v2f*)(&qs[4 * (c + 1) + 2 * hi]);
            v2f b1 = *(const v2f*)(krow + 4 * (c + 1));
            c1 = __builtin_amdgcn_wmma_f32_16x16x4_f32(
                     false, a1, false, b1, (short)0, c1, false, false);
        }
        // D[m][n] identical over m (A rows identical): lane L holds score of key (L&15).
        float score = (c0[0] + c1[0]) * scale;

        const int key = blk * BLOCK_SZ + lo;
        score += slope * (float)(key - qpos);
        if (key >= ctx) score = -1e30f;

        // max over the 16 keys (halves are duplicates; xor-reduce within 16 lanes)
        float bm = score;
#pragma unroll
        for (int off = 8; off >= 1; off >>= 1)
            bm = fmaxf(bm, __shfl_xor(bm, off, 32));

        const float new_m = fmaxf(run_m, bm);
        const float corr  = __expf(run_m - new_m);   // 0 on first block (run_m=-inf)
        const float p     = __expf(score - new_m);   // masked keys -> 0

        float psum = p;
#pragma unroll
        for (int off = 8; off >= 1; off >>= 1)
            psum += __shfl_xor(psum, off, 32);

        run_l = run_l * corr + psum;
        run_m = new_m;
        acc.x *= corr; acc.y *= corr; acc.z *= corr; acc.w *= corr;

        // ---- acc += sum_j p[j] * V[j][:]  (fully coalesced b128 row streams) ----
        const float* vbase = v_cache + kvbase + 4 * lane;
#pragma unroll
        for (int j = 0; j < BLOCK_SZ; ++j) {
            const float pj = __shfl(p, j, 32);                 // weight of key j
            v4f v = *(const v4f*)(vbase + (size_t)j * HEAD_DIM);
            acc.x += pj * v.x; acc.y += pj * v.y;
            acc.z += pj * v.z; acc.w += pj * v.w;
        }
    }

    // ---- cross-wave flash combine through LDS ----
    if (lane == 0) { red_m[wave] = run_m; red_l[wave] = run_l; }
    *(v4f*)(&red_acc[wave][4 * lane]) = acc;
    __syncthreads();

    if (tid < HEAD_DIM) {
        float M = -INFINITY;
#pragma unroll
        for (int w = 0; w < WAVES; ++w) M = fmaxf(M, red_m[w]);
        float L = 0.0f, o = 0.0f;
#pragma unroll
        for (int w = 0; w < WAVES; ++w) {
            const float f = __expf(red_m[w] - M);   // 0 for waves that did no work
            L += red_l[w] * f;
            o += red_acc[w][tid] * f;
        }
        out[((size_t)s * N_HEAD + h) * HEAD_DIM + tid] = o / L;
    }
}

extern "C" void kernel_launch(void* const* d_in, const int* in_sizes, int n_in,
                              void* d_out, int out_size, void* d_ws, size_t ws_size,
                              hipStream_t stream) {
    const float* q            = (const float*)d_in[0];
    const float* k_cache      = (const float*)d_in[1];
    const float* v_cache      = (const float*)d_in[2];
    const int*   block_tables = (const int*)d_in[3];
    const int*   context_lens = (const int*)d_in[4];
    float*       out          = (float*)d_out;

    dim3 grid(N_HEAD, NUM_SEQS);
    dim3 block(NTHREADS);
    paged_attn_alibi_wmma<<<grid, block, 0, stream>>>(
        q, k_cache, v_cache, block_tables, context_lens, out);
}